// YOLOLoss_30348238913982
// MI455X (gfx1250) — compile-verified
//
#include <hip/hip_runtime.h>
#include <math.h>

#define BN 16
#define NN 16800
#define CC 80
#define MM 32
#define ALPHA 0.25f
#define THR 0.25f
#define BLOCK 256
#define NBLK_PER_IMG ((NN + BLOCK - 1) / BLOCK)   /* 66 */
#define NBLK (BN * NBLK_PER_IMG)                  /* 1056 */

typedef float v2f __attribute__((ext_vector_type(2)));
typedef float v8f __attribute__((ext_vector_type(8)));

struct ImgInfo { int use_fb; float w; int fb[5]; int pad; };  // 32 bytes

// ---------- math helpers (mirror reference exactly) ----------

__device__ __forceinline__ float4 sanitize_box(float4 b) {
  float v[4] = {b.x, b.y, b.z, b.w};
#pragma unroll
  for (int i = 0; i < 4; ++i) {
    float x = v[i];
    if (isnan(x)) x = 0.0f;
    else if (isinf(x)) x = (x > 0.0f) ? 1e4f : -1e4f;
    v[i] = x;
  }
  float x1 = fminf(v[0], v[2]);
  float y1 = fminf(v[1], v[3]);
  float x2 = fmaxf(v[0], v[2]);
  float y2 = fmaxf(v[1], v[3]);
  x2 = fmaxf(x2, x1 + 1e-6f);
  y2 = fmaxf(y2, y1 + 1e-6f);
  return make_float4(x1, y1, x2, y2);
}

__device__ __forceinline__ float giou(float4 a, float4 b) {
  float area_a = (a.z - a.x) * (a.w - a.y);
  float area_b = (b.z - b.x) * (b.w - b.y);
  float ltx = fmaxf(a.x, b.x), lty = fmaxf(a.y, b.y);
  float rbx = fminf(a.z, b.z), rby = fminf(a.w, b.w);
  float iw = fmaxf(rbx - ltx, 0.0f), ih = fmaxf(rby - lty, 0.0f);
  float inter = iw * ih;
  float uni = area_a + area_b - inter;
  float iou = inter / uni;
  float cltx = fminf(a.x, b.x), clty = fminf(a.y, b.y);
  float crbx = fmaxf(a.z, b.z), crby = fmaxf(a.w, b.w);
  float cw = fmaxf(crbx - cltx, 0.0f), ch = fmaxf(crby - clty, 0.0f);
  float carea = cw * ch;
  return iou - (carea - uni) / carea;
}

__device__ __forceinline__ float log_sigmoid(float x) {
  return fminf(x, 0.0f) - log1pf(expf(-fabsf(x)));
}
__device__ __forceinline__ float bce_logits(float x, float t) {
  return -(t * log_sigmoid(x) + (1.0f - t) * log_sigmoid(-x));
}
__device__ __forceinline__ float focal_term(float x, int is_target) {
  float t = is_target ? 1.0f : 0.0f;
  float ce = bce_logits(x, t);
  float p = 1.0f / (1.0f + expf(-x));
  float p_t = p * t + (1.0f - p) * (1.0f - t);
  float a_t = ALPHA * t + (1.0f - ALPHA) * (1.0f - t);
  float om = 1.0f - p_t;
  return a_t * om * om * ce;
}

// ---------- WMMA wave32 reduction: D = A(16x4) x ones(4x16) ----------
// Lane l puts v in A K-slot {0 (l<16), 2 (l>=16)}; D[m][n] = v_m + v_{m+16}.
// Lanes 0-15 hold rows 0..7 in d[0..7]; lanes 16-31 hold rows 8..15.
// Sum of d[0..7] + shfl_xor(16) => full 32-lane sum in every lane.
__device__ __forceinline__ float wave_sum_wmma(float v) {
  v2f a; a.x = v;    a.y = 0.0f;
  v2f o; o.x = 1.0f; o.y = 1.0f;
  v8f c = {0.0f, 0.0f, 0.0f, 0.0f, 0.0f, 0.0f, 0.0f, 0.0f};
  c = __builtin_amdgcn_wmma_f32_16x16x4_f32(false, a, false, o,
                                            (short)0, c, false, false);
  float s = c[0] + c[1] + c[2] + c[3] + c[4] + c[5] + c[6] + c[7];
  s += __shfl_xor(s, 16, 32);
  return s;
}

// ---------- kernels ----------

__global__ void k_init(int* __restrict__ pos_count) {
  int i = threadIdx.x;
  if (i < BN) pos_count[i] = 0;
}

__global__ __launch_bounds__(BLOCK)
void k_match(const float* __restrict__ box_preds,
             const float* __restrict__ gt_boxes,
             float* __restrict__ best_iou,
             int* __restrict__ best_idx,
             int* __restrict__ pos_count) {
  __shared__ float4 sgt[MM];
  int b  = blockIdx.x / NBLK_PER_IMG;
  int nb = blockIdx.x % NBLK_PER_IMG;
  int n  = nb * BLOCK + threadIdx.x;
  if (threadIdx.x < MM) {
    const float4* g = (const float4*)(gt_boxes) + (size_t)b * MM;
    sgt[threadIdx.x] = sanitize_box(g[threadIdx.x]);
  }
  __syncthreads();
  if (n >= NN) return;
  float4 p = sanitize_box(((const float4*)box_preds)[(size_t)b * NN + n]);
  float best = -1e30f; int bi = 0;
#pragma unroll 4
  for (int m = 0; m < MM; ++m) {
    float g = giou(p, sgt[m]);
    if (g > best) { best = g; bi = m; }   // strict > keeps first max (jnp.argmax)
  }
  size_t o = (size_t)b * NN + n;
  best_iou[o] = best;
  best_idx[o] = bi;
  if (best > THR) atomicAdd(&pos_count[b], 1);  // int atomic: deterministic
}

__global__ __launch_bounds__(BLOCK)
void k_select(const float* __restrict__ best_iou,
              const int* __restrict__ pos_count,
              ImgInfo* __restrict__ info) {
  int b = blockIdx.x;
  int t = threadIdx.x;
  __shared__ float sval[BLOCK * 5];
  __shared__ int   sidx[BLOCK * 5];

  if (pos_count[b] > 0) {
    if (t == 0) {
      info[b].use_fb = 0; info[b].w = 1.0f;
      for (int j = 0; j < 5; ++j) info[b].fb[j] = -1;
    }
    return;  // uniform branch per block
  }
  // Stable per-thread top-5 of clip(best_iou, 0) (ties -> lower index first,
  // guaranteed by ascending scan + strict-> insertion).
  float lv[5] = {-1.0f, -1.0f, -1.0f, -1.0f, -1.0f};
  int   li[5] = {-1, -1, -1, -1, -1};
  for (int n = t; n < NN; n += BLOCK) {
    float v = fmaxf(best_iou[(size_t)b * NN + n], 0.0f);
    if (v > lv[4]) {
      int j = 4;
      while (j > 0 && v > lv[j - 1]) { lv[j] = lv[j - 1]; li[j] = li[j - 1]; --j; }
      lv[j] = v; li[j] = n;
    }
  }
  for (int j = 0; j < 5; ++j) { sval[t * 5 + j] = lv[j]; sidx[t * 5 + j] = li[j]; }
  __syncthreads();
  if (t == 0) {
    int sel[5]; float selv[5];
    for (int k = 0; k < 5; ++k) {
      float bv = -2.0f; int bidx = 0x7fffffff; int bslot = -1;
      for (int s = 0; s < BLOCK * 5; ++s) {
        int idx = sidx[s];
        if (idx < 0) continue;
        float v = sval[s];
        if (v > bv || (v == bv && idx < bidx)) { bv = v; bidx = idx; bslot = s; }
      }
      sel[k] = bidx; selv[k] = bv;
      if (bslot >= 0) sidx[bslot] = -1;
    }
    int any = 0;
    for (int k = 0; k < 5; ++k) {
      if (selv[k] > 0.0f) { info[b].fb[k] = sel[k]; any = 1; }
      else                info[b].fb[k] = -1;
    }
    info[b].use_fb = 1;
    info[b].w = any ? 1.0f : 0.0f;
  }
}

__global__ __launch_bounds__(BLOCK)
void k_main(const float* __restrict__ cls_preds,
            const float* __restrict__ obj_preds,
            const int* __restrict__ gt_labels,
            const float* __restrict__ best_iou,
            const int* __restrict__ best_idx,
            const ImgInfo* __restrict__ info,
            float* __restrict__ partials) {
  int b  = blockIdx.x / NBLK_PER_IMG;
  int nb = blockIdx.x % NBLK_PER_IMG;
  int n  = nb * BLOCK + threadIdx.x;

  float s_box = 0.0f, s_cls = 0.0f, s_obj = 0.0f, s_np = 0.0f, s_nn = 0.0f;
  ImgInfo ii = info[b];

  if (n < NN && ii.w != 0.0f) {
    size_t o = (size_t)b * NN + n;
    float best = best_iou[o];
    bool pos;
    if (ii.use_fb)
      pos = (n == ii.fb[0]) | (n == ii.fb[1]) | (n == ii.fb[2]) |
            (n == ii.fb[3]) | (n == ii.fb[4]);
    else
      pos = best > THR;
    float posf = pos ? 1.0f : 0.0f;
    float negf = (best < 0.5f) ? 1.0f : 0.0f;

    s_box = posf * (1.0f - best);       // giou_diag == best_iou (same arithmetic)

    float x = obj_preds[o];
    float bce = bce_logits(x, posf);
    s_obj = posf * bce + 0.5f * negf * bce;
    s_np = posf; s_nn = negf;

    if (pos) {  // posf==0 -> cls contribution exactly 0: skip the 320B read
      int lab = gt_labels[b * MM + best_idx[o]];
      const float4* cp = (const float4*)(cls_preds + o * CC);
      float acc = 0.0f;
#pragma unroll 4
      for (int c4 = 0; c4 < CC / 4; ++c4) {
        float4 l = cp[c4];
        int c = c4 * 4;
        acc += focal_term(l.x, c + 0 == lab);
        acc += focal_term(l.y, c + 1 == lab);
        acc += focal_term(l.z, c + 2 == lab);
        acc += focal_term(l.w, c + 3 == lab);
      }
      s_cls = acc;
    }
  }

  // Deterministic block reduction: WMMA per-wave sum, then LDS across 8 waves.
  __shared__ float red[8 * 5];
  int wave = threadIdx.x >> 5;
  int lane = threadIdx.x & 31;
  float vals[5] = {s_box, s_cls, s_obj, s_np, s_nn};
#pragma unroll
  for (int q = 0; q < 5; ++q) {
    float wsum = wave_sum_wmma(vals[q]);  // EXEC all-1s here (reconverged)
    if (lane == 0) red[wave * 5 + q] = wsum;
  }
  __syncthreads();
  if (threadIdx.x == 0) {
#pragma unroll
    for (int q = 0; q < 5; ++q) {
      float tsum = 0.0f;
      for (int w = 0; w < 8; ++w) tsum += red[w * 5 + q];
      partials[(size_t)blockIdx.x * 5 + q] = tsum;
    }
  }
}

__global__ void k_final(const float* __restrict__ partials,
                        float* __restrict__ out) {
  if (threadIdx.x != 0) return;
  float box = 0.0f, cls = 0.0f, obj = 0.0f, np = 0.0f, nn = 0.0f;
  for (int i = 0; i < NBLK; ++i) {
    box += partials[i * 5 + 0];
    cls += partials[i * 5 + 1];
    obj += partials[i * 5 + 2];
    np  += partials[i * 5 + 3];
    nn  += partials[i * 5 + 4];
  }
  float main_v = 5.0f * box / np + 1.0f * cls / np + obj / fmaxf(np + nn, 1.0f);
  float fb = ((nn > 0.0f) && (obj > 0.0f)) ? 0.1f * obj / fmaxf(nn, 1.0f) : 0.01f;
  out[0] = (np > 0.0f) ? main_v : fb;
}

// ---------- launch ----------

extern "C" void kernel_launch(void* const* d_in, const int* in_sizes, int n_in,
                              void* d_out, int out_size, void* d_ws, size_t ws_size,
                              hipStream_t stream) {
  const float* box_preds = (const float*)d_in[0];
  const float* cls_preds = (const float*)d_in[1];
  const float* obj_preds = (const float*)d_in[2];
  const float* gt_boxes  = (const float*)d_in[3];
  const int*   gt_labels = (const int*)d_in[4];

  char* ws = (char*)d_ws;
  float*   best_iou  = (float*)ws;   ws += sizeof(float) * (size_t)BN * NN;
  int*     best_idx  = (int*)ws;     ws += sizeof(int)   * (size_t)BN * NN;
  int*     pos_count = (int*)ws;     ws += sizeof(int)   * 64;            // padded
  ImgInfo* info      = (ImgInfo*)ws; ws += sizeof(ImgInfo) * BN;
  float*   partials  = (float*)ws;   /* NBLK*5 floats */

  k_init  <<<1,    64,    0, stream>>>(pos_count);
  k_match <<<NBLK, BLOCK, 0, stream>>>(box_preds, gt_boxes, best_iou, best_idx, pos_count);
  k_select<<<BN,   BLOCK, 0, stream>>>(best_iou, pos_count, info);
  k_main  <<<NBLK, BLOCK, 0, stream>>>(cls_preds, obj_preds, gt_labels,
                                       best_iou, best_idx, info, partials);
  k_final <<<1,    64,    0, stream>>>(partials, (float*)d_out);
}